// GetCostVolume_26465588478488
// MI455X (gfx1250) — compile-verified
//
#include <hip/hip_runtime.h>

typedef float v2f __attribute__((ext_vector_type(2)));
typedef float v4f __attribute__((ext_vector_type(4)));
typedef float v8f __attribute__((ext_vector_type(8)));

#define BB 2
#define CC 32
#define HH 128
#define WW 256
#define DD 48

#define XW 272      // x pair-row width (float2 units); 2*272 % 64 == 32 -> disjoint bank halves
#define YW 336      // 64 left pad + 256 + 16 right pad; 2*336 % 64 == 32
#define YPADL 64
#define CORRW 100   // band width (need 81; WMMA writes 96)
#define CORRM 20    // padded inner (M) dim: 20*4B=80B keeps b128 16B-aligned, banks spread

// cost[b,d,h,w] = (1/C) * [ (1-fx)*corr(w,x0)*v0 + fx*corr(w,x0+1)*v1 ],
// corr(w,w') = sum_c x[b,c,h,w]*y[b,c,h,w'], banded V_WMMA_F32_16X16X4_F32 GEMM.
__global__ __launch_bounds__(256)
void cost_volume_wmma_kernel(const float* __restrict__ x,
                             const float* __restrict__ y,
                             const float* __restrict__ disp,
                             float* __restrict__ out)
{
    extern __shared__ char smemraw[];
    v2f*   xp2  = (v2f*)smemraw;                    // [16 ch-pairs][XW]
    v2f*   yp2  = xp2 + 16 * XW;                    // [16 ch-pairs][YW]; col = w' + 64
    float* corr = (float*)(yp2 + 16 * YW);          // [16 wtiles][CORRW cols][CORRM m]

    const int tid = threadIdx.x;
    const int b = blockIdx.x / HH;
    const int h = blockIdx.x % HH;

    const float* xrow = x + ((size_t)(b * CC) * HH + h) * WW;
    const float* yrow = y + ((size_t)(b * CC) * HH + h) * WW;

    // ---- stage X / zero-padded Y rows into LDS as channel pairs (b64 stores) ----
    for (int idx = tid; idx < 16 * WW; idx += 256) {
        int p = idx >> 8;                // channel pair
        int w = idx & (WW - 1);
        v2f xv, yv;
        xv.x = xrow[(size_t)(2 * p + 0) * HH * WW + w];
        xv.y = xrow[(size_t)(2 * p + 1) * HH * WW + w];
        yv.x = yrow[(size_t)(2 * p + 0) * HH * WW + w];
        yv.y = yrow[(size_t)(2 * p + 1) * HH * WW + w];
        xp2[p * XW + w]         = xv;
        yp2[p * YW + YPADL + w] = yv;
    }
    for (int idx = tid; idx < 16 * 80; idx += 256) {  // 64 left + 16 right pad per pair
        int p = idx / 80;
        int q = idx % 80;
        int col = (q < YPADL) ? q : (YPADL + WW + (q - YPADL));
        yp2[p * YW + col] = (v2f){0.0f, 0.0f};
    }
    __syncthreads();

    // ---- banded correlation via V_WMMA_F32_16X16X4_F32 ----
    const int lane = tid & 31;
    const int wave = tid >> 5;
    const int l    = lane & 15;
    const int half = lane >> 4;          // 0: K={0,1}/M=r ; 1: K={2,3}/M=r+8

    for (int i = wave; i < 16; i += 8) { // w-tile index (uniform per wave)
        // Preload all 8 A fragments (one b64 each) for this w-tile.
        v2f a[8];
        const int m = 16 * i + l;
#pragma unroll
        for (int kc = 0; kc < 8; ++kc)
            a[kc] = xp2[(2 * kc + half) * XW + m];

#pragma unroll
        for (int jidx = 0; jidx < 6; ++jidx) {           // band w' in [16i-64, 16i+32)
            const int wp = 16 * (i - 4 + jidx) + l + YPADL;
            // Batch-load all 8 B fragments, then chain the WMMAs.
            v2f bf[8];
#pragma unroll
            for (int kc = 0; kc < 8; ++kc)
                bf[kc] = yp2[(2 * kc + half) * YW + wp];

            v8f acc = {0.f, 0.f, 0.f, 0.f, 0.f, 0.f, 0.f, 0.f};
#pragma unroll
            for (int kc = 0; kc < 8; ++kc)
                acc = __builtin_amdgcn_wmma_f32_16x16x4_f32(
                          false, a[kc], false, bf[kc], (short)0, acc, false, false);

            // D layout: VGPR r -> M = r + 8*half, N = l. Store as two b128s
            // into transposed band corr[i][col=16*jidx+l][m].
            float* cb = corr + ((size_t)(i * CORRW + 16 * jidx + l) * CORRM + 8 * half);
            *(v4f*)(cb + 0) = __builtin_shufflevector(acc, acc, 0, 1, 2, 3);
            *(v4f*)(cb + 4) = __builtin_shufflevector(acc, acc, 4, 5, 6, 7);
        }
    }
    __syncthreads();

    // ---- gather + lerp epilogue ----
    const float* drow = disp + ((size_t)(b * DD) * HH + h) * WW;
    float*       orow = out  + ((size_t)(b * DD) * HH + h) * WW;
    for (int e = tid; e < DD * WW; e += 256) {
        int d = e >> 8;
        int w = e & (WW - 1);
        float dv  = drow[(size_t)d * HH * WW + w];
        float px  = (float)w - dv;
        float x0f = floorf(px);
        float fx  = px - x0f;
        int   x0  = (int)x0f;
        int   i   = w >> 4;
        int   col0 = x0 - 16 * i + 64;            // band column of x0
        int   m    = w & 15;
        float w0 = (x0     >= 0 && x0     < WW) ? (1.0f - fx) : 0.0f;
        float w1 = (x0 + 1 >= 0 && x0 + 1 < WW) ? fx          : 0.0f;
        int c0 = min(max(col0,     0), CORRW - 1);
        int c1 = min(max(col0 + 1, 0), CORRW - 1);
        float v0 = corr[(size_t)(i * CORRW + c0) * CORRM + m];
        float v1 = corr[(size_t)(i * CORRW + c1) * CORRM + m];
        orow[(size_t)d * HH * WW + w] = (v0 * w0 + v1 * w1) * (1.0f / CC);
    }
}

extern "C" void kernel_launch(void* const* d_in, const int* in_sizes, int n_in,
                              void* d_out, int out_size, void* d_ws, size_t ws_size,
                              hipStream_t stream) {
    const float* x    = (const float*)d_in[0];
    const float* y    = (const float*)d_in[1];
    const float* disp = (const float*)d_in[2];
    float* out = (float*)d_out;

    const size_t smem = (size_t)(16 * XW * 2 + 16 * YW * 2 + 16 * CORRW * CORRM)
                        * sizeof(float);   // 205824 B, within CDNA5's 320KB WGP LDS
    (void)hipFuncSetAttribute((const void*)cost_volume_wmma_kernel,
                              hipFuncAttributeMaxDynamicSharedMemorySize, (int)smem);

    cost_volume_wmma_kernel<<<dim3(BB * HH), dim3(256), smem, stream>>>(x, y, disp, out);
}